// Real3DDecoder_15719580304115
// MI455X (gfx1250) — compile-verified
//
#include <hip/hip_runtime.h>
#include <math.h>
#include <stdint.h>

// Problem constants (fixed by setup_inputs)
#define B_   16
#define C_   20
#define H_   192
#define W_   480
#define HW_  (H_ * W_)
#define K_   500
#define PI_  3.14159265358979323846f
#define FOCAL_ 721.5377f

// ---- CDNA5 async-copy primitives -------------------------------------------
// Wait until ASYNCcnt <= n (split wait counters, gfx1250).
#if __has_builtin(__builtin_amdgcn_s_wait_asynccnt)
#define ASYNC_WAIT(n) __builtin_amdgcn_s_wait_asynccnt(n)
#else
#define ASYNC_WAIT(n) asm volatile("s_wait_asynccnt " #n ::: "memory")
#endif

// One 4-byte async copy global -> LDS per lane (GLOBAL_LOAD_ASYNC_TO_LDS_B32,
// GV mode: 64-bit address in VGPR pair, LDS byte offset in VDST VGPR).
__device__ __forceinline__ void async_copy_b32(uint32_t lds_off, const float* gptr) {
  asm volatile("global_load_async_to_lds_b32 %0, %1, off"
               :: "v"(lds_off), "v"(gptr)
               : "memory");
}

__device__ __forceinline__ float sigmoidf_(float x) {
  return 1.0f / (1.0f + expf(-x));
}

__device__ __forceinline__ float fmt_ang(float a) {
  a = (a >  PI_) ? a - 2.0f * PI_ : a;
  a = (a < -PI_) ? a + 2.0f * PI_ : a;
  return a;
}

// ---- Kernel 1: sigmoid + 3x3 peak NMS + class-max score --------------------
// Tile 32x8 pixels per 256-thread block, halo 34x10 per channel, async
// double-buffered in LDS across the 20 channels.
__global__ __launch_bounds__(256)
void score_nms_kernel(const float* __restrict__ hm, float* __restrict__ scores) {
  const int tid = threadIdx.x;
  const int lx  = tid & 31;           // 0..31
  const int ly  = tid >> 5;           // 0..7
  const int bx0 = blockIdx.x * 32;
  const int by0 = blockIdx.y * 8;
  const int b   = blockIdx.z;
  const int x   = bx0 + lx;
  const int y   = by0 + ly;

  __shared__ float buf[2][512];       // 34x10 = 340 halo cells, padded to 512

  auto issue = [&](int c, int bb) {
    const float* base = hm + ((size_t)(b * C_ + c)) * HW_;
    // cell tid (all 256 threads) -- 1st async instruction per wave
    {
      int cell = tid;
      int hy = cell / 34, hx = cell - hy * 34;
      int gy = min(max(by0 - 1 + hy, 0), H_ - 1);
      int gx = min(max(bx0 - 1 + hx, 0), W_ - 1);
      uint32_t off = (uint32_t)(size_t)&buf[bb][cell];
      async_copy_b32(off, base + gy * W_ + gx);
    }
    // cell 256+tid (cells >=340 are dummy pads so every wave issues exactly 2)
    {
      int cell = 256 + tid;
      int src  = (cell < 340) ? cell : 339;
      int hy = src / 34, hx = src - hy * 34;
      int gy = min(max(by0 - 1 + hy, 0), H_ - 1);
      int gx = min(max(bx0 - 1 + hx, 0), W_ - 1);
      uint32_t off = (uint32_t)(size_t)&buf[bb][cell];
      async_copy_b32(off, base + gy * W_ + gx);
    }
  };

  issue(0, 0);
  float best = 0.0f;

  for (int c = 0; c < C_; ++c) {
    const int cur = c & 1;
    if (c + 1 < C_) { issue(c + 1, cur ^ 1); ASYNC_WAIT(2); }  // ch c done (in-order)
    else            { ASYNC_WAIT(0); }
    __syncthreads();

    const float* t = buf[cur];
    const float v = t[(ly + 1) * 34 + (lx + 1)];
    float m = -INFINITY;
#pragma unroll
    for (int dy = -1; dy <= 1; ++dy) {
      const int gy = y + dy;
      if ((unsigned)gy < (unsigned)H_) {
#pragma unroll
        for (int dx = -1; dx <= 1; ++dx) {
          const int gx = x + dx;
          if ((unsigned)gx < (unsigned)W_)
            m = fmaxf(m, t[(ly + 1 + dy) * 34 + (lx + 1 + dx)]);
        }
      }
    }
    // sigmoid is monotone: raw-logit window max <=> sigmoid window max
    if (v == m) best = fmaxf(best, sigmoidf_(v));
    __syncthreads();                  // readers done before next issue overwrites
  }

  scores[(size_t)b * HW_ + y * W_ + x] = best;
}

// ---- Kernel 2: per-batch radix-select top-500 + bitonic sort + decode ------
__global__ __launch_bounds__(1024)
void topk_decode_kernel(const float* __restrict__ scores,
                        const float* __restrict__ dep,
                        const float* __restrict__ wh,
                        const float* __restrict__ rot,
                        const float* __restrict__ dim,
                        const float* __restrict__ calib,
                        const int*   __restrict__ ori,
                        float* __restrict__ out) {
  const int b   = blockIdx.x;
  const int tid = threadIdx.x;
  const int NT  = blockDim.x;         // 1024
  const float* S = scores + (size_t)b * HW_;

  __shared__ unsigned hist[256];
  __shared__ float    sVal[512];
  __shared__ int      sIdx[512];
  __shared__ unsigned sPrefix, sRemK, cntGT, cntEq;

  if (tid == 0) { sPrefix = 0u; sRemK = K_; }
  __syncthreads();

  // 4-pass MSB-first radix select. Scores >= 0 so float bits order == value order.
  for (int p = 3; p >= 0; --p) {
    if (tid < 256) hist[tid] = 0u;
    __syncthreads();
    const unsigned pref  = sPrefix;
    const unsigned hmask = (p == 3) ? 0u : (0xFFFFFFFFu << (8 * (p + 1)));
    for (int i = tid; i < HW_; i += NT) {
      const unsigned v = __float_as_uint(S[i]);
      if ((v & hmask) == pref)
        atomicAdd(&hist[(v >> (8 * p)) & 255u], 1u);
    }
    __syncthreads();
    if (tid == 0) {
      unsigned K = sRemK, cum = 0; int sel = 0;
      for (int bin = 255; bin >= 0; --bin) {
        const unsigned cnt = hist[bin];
        if (cum + cnt >= K) { sel = bin; sRemK = K - cum; break; }
        cum += cnt;
      }
      sPrefix = pref | ((unsigned)sel << (8 * p));
    }
    __syncthreads();
  }

  const unsigned T      = sPrefix;        // exact bit pattern of 500th value
  const unsigned needEq = sRemK;          // how many == T we must take
  const unsigned nGT    = K_ - needEq;    // count strictly greater than T

  if (tid == 0) { cntGT = 0u; cntEq = 0u; }
  if (tid < 512) { sVal[tid] = -1.0f; sIdx[tid] = 0x7FFFFFFF; }
  __syncthreads();

  for (int i = tid; i < HW_; i += NT) {
    const unsigned v = __float_as_uint(S[i]);
    if (v > T) {
      const unsigned pos = atomicAdd(&cntGT, 1u);
      sVal[pos] = S[i]; sIdx[pos] = i;
    } else if (v == T) {
      const unsigned pos = atomicAdd(&cntEq, 1u);
      if (pos < needEq) { sVal[nGT + pos] = S[i]; sIdx[nGT + pos] = i; }
    }
  }
  __syncthreads();

  // Bitonic sort of 512 entries: value descending, index ascending on ties
  for (int k = 2; k <= 512; k <<= 1) {
    for (int j = k >> 1; j > 0; j >>= 1) {
      if (tid < 512) {
        const int i = tid, ixj = i ^ j;
        if (ixj > i) {
          const float vi = sVal[i], vx = sVal[ixj];
          const int   ii = sIdx[i], ix = sIdx[ixj];
          const bool iFirst = (vi > vx) || (vi == vx && ii < ix);
          const bool up = ((i & k) == 0);
          if (up ? !iFirst : iFirst) {
            sVal[i] = vx; sVal[ixj] = vi;
            sIdx[i] = ix; sIdx[ixj] = ii;
          }
        }
      }
      __syncthreads();
    }
  }

  // Per-detection geometry decode
  if (tid < K_) {
    const int   k     = tid;
    const float score = sVal[k];
    const int   idx   = sIdx[k];
    const float u = (float)(idx % W_);
    const float v = (float)(idx / W_);

    const float* cal = calib + b * 12;
    const float fx = cal[0], cx = cal[2], tx = cal[3];
    const float fy = cal[5], cy = cal[6], ty = cal[7];
    const float c23 = cal[11];
    const float oh = (float)ori[b * 2 + 0];
    const float ow = (float)ori[b * 2 + 1];
    const float r  = fmaxf(ow, oh) / (float)W_;
    const float tX = ow * 0.5f - r * (float)W_ * 0.5f;
    const float tY = oh * 0.5f - r * (float)H_ * 0.5f;

    const float d0   = dep[(size_t)b * HW_ + idx];
    const float depm = 1.0f / (sigmoidf_(d0) + 1e-6f) - 1.0f;
    const float wh0  = wh[((size_t)b * 2 + 0) * HW_ + idx];
    const float wh1  = wh[((size_t)b * 2 + 1) * HW_ + idx];
    const float dim0 = dim[((size_t)b * 3 + 0) * HW_ + idx];
    float rg[12];
#pragma unroll
    for (int c = 0; c < 12; ++c) rg[c] = rot[((size_t)b * 12 + c) * HW_ + idx];

    const float x1 = (u - wh0 * 0.5f) * r + tX;
    const float y1 = (v - wh1 * 0.5f) * r + tY;
    const float x2 = (u + wh0 * 0.5f) * r + tX;
    const float y2 = (v + wh1 * 0.5f) * r + tY;
    const float cox = u * r + tX;
    const float coy = v * r + tY;

    const float scaler = (fx + fy) * 0.5f / FOCAL_;
    const float dep_s  = depm * scaler;
    const float z = dep_s - c23;
    const float X = (cox * dep_s - tx - cx * z) / fx;
    const float Y = (coy * dep_s - ty - cy * z) / fy;
    const float locY = Y + dim0 * 0.5f;

    int bin = 0; float bv = rg[0];
#pragma unroll
    for (int j = 1; j < 4; ++j) if (rg[j] > bv) { bv = rg[j]; bin = j; }  // first-max
    float s = rg[4 + 2 * bin], cc = rg[5 + 2 * bin];
    const float n = fmaxf(sqrtf(s * s + cc * cc), 1e-12f);
    s /= n; cc /= n;
    const float binc[4] = {0.0f, PI_ * 0.5f, PI_, -PI_ * 0.5f};
    const float az = fmt_ang(atan2f(s, cc) + binc[bin]);
    const float ax = fmt_ang(az - PI_ * 0.5f);
    const float th = fmt_ang(-atan2f(X, z));
    const float ry = fmt_ang(ax - th);

    float* o = out + ((size_t)b * K_ + k) * 13;
    o[0] = score;
    o[1] = x1; o[2] = y1; o[3] = x2; o[4] = y2;
    o[5] = cox; o[6] = coy;
    o[7] = dep_s;
    o[8] = X; o[9] = locY; o[10] = z;
    o[11] = ax; o[12] = ry;
  }
}

extern "C" void kernel_launch(void* const* d_in, const int* in_sizes, int n_in,
                              void* d_out, int out_size, void* d_ws, size_t ws_size,
                              hipStream_t stream) {
  const float* hm  = (const float*)d_in[0];
  const float* dep = (const float*)d_in[1];
  const float* wh  = (const float*)d_in[2];
  const float* rot = (const float*)d_in[3];
  const float* dim = (const float*)d_in[4];
  const float* cal = (const float*)d_in[5];
  const int*   ori = (const int*)d_in[6];
  float* out    = (float*)d_out;
  float* sc_ws  = (float*)d_ws;       // B*HW floats = 5.9 MB scratch

  dim3 gA(W_ / 32, H_ / 8, B_);       // 15 x 24 x 16 blocks
  score_nms_kernel<<<gA, 256, 0, stream>>>(hm, sc_ws);
  topk_decode_kernel<<<B_, 1024, 0, stream>>>(sc_ws, dep, wh, rot, dim, cal, ori, out);
}